// QuantumHybridClassifier_65481071406539
// MI455X (gfx1250) — compile-verified
//
#include <hip/hip_runtime.h>
#include <math.h>

typedef __attribute__((ext_vector_type(2))) float v2f;
typedef __attribute__((ext_vector_type(8))) float v8f;

// ---------------- problem sizes ----------------
#define BATCH 64
// conv1: [64,3,250,250] -> [64,6,124,124] (k=5, s=2, p=1)
#define SZ_CONV1 (BATCH * 6 * 124 * 124)        // 5,904,384
// pool1 -> [64,6,123,123]
#define SZ_POOL1 (BATCH * 6 * 123 * 123)        // 5,809,536
// conv2: -> [64,15,62,62] (k=3, s=2, p=1)
#define SZ_CONV2 (BATCH * 15 * 62 * 62)         // 3,690,240
// pool2 -> [64,15,61,61] flattened [64,55815], padded stride 55816
#define K1   55815
#define LDX  55816                              // multiple of 4 -> aligned v2f loads
#define N1   120
#define K2   120
#define N2   84
#define SPLITS 56                               // split-K blocks per fc1 tile
#define S_STEPS ((K1 + 3) / 4)                  // 13954 WMMA K-steps (last is partial)

// =============================================================
// conv1 + ReLU (direct, weights in LDS)
// =============================================================
__global__ __launch_bounds__(256) void conv1_relu_kernel(
    const float* __restrict__ in, const float* __restrict__ w,
    const float* __restrict__ bias, float* __restrict__ out)
{
    __shared__ float sw[6 * 3 * 25];   // 450
    __shared__ float sb[6];
    for (int i = threadIdx.x; i < 450; i += 256) sw[i] = w[i];
    if (threadIdx.x < 6) sb[threadIdx.x] = bias[threadIdx.x];
    __syncthreads();

    int tid = blockIdx.x * 256 + threadIdx.x;
    if (tid >= SZ_CONV1) return;
    int ox = tid % 124; int t = tid / 124;
    int oy = t % 124;   t /= 124;
    int oc = t % 6;     int b = t / 6;

    const float* inb = in + (b * 3) * 250 * 250;
    const float* wf  = sw + oc * 75;
    float acc = sb[oc];
    int iy0 = oy * 2 - 1, ix0 = ox * 2 - 1;
#pragma unroll
    for (int ic = 0; ic < 3; ++ic) {
        const float* inc = inb + ic * 250 * 250;
#pragma unroll
        for (int ky = 0; ky < 5; ++ky) {
            int iy = iy0 + ky;
            if (iy < 0 || iy >= 250) continue;
            const float* row = inc + iy * 250;
#pragma unroll
            for (int kx = 0; kx < 5; ++kx) {
                int ix = ix0 + kx;
                if (ix >= 0 && ix < 250)
                    acc = fmaf(row[ix], wf[ic * 25 + ky * 5 + kx], acc);
            }
        }
    }
    out[tid] = fmaxf(acc, 0.0f);
}

// =============================================================
// maxpool 2x2 stride 1 : [64,6,124,124] -> [64,6,123,123]
// =============================================================
__global__ __launch_bounds__(256) void pool1_kernel(
    const float* __restrict__ in, float* __restrict__ out)
{
    int tid = blockIdx.x * 256 + threadIdx.x;
    if (tid >= SZ_POOL1) return;
    int x = tid % 123; int t = tid / 123;
    int y = t % 123;   t /= 123;
    int c = t % 6;     int b = t / 6;
    const float* p = in + ((b * 6 + c) * 124 + y) * 124 + x;
    out[tid] = fmaxf(fmaxf(p[0], p[1]), fmaxf(p[124], p[125]));
}

// =============================================================
// conv2 + ReLU : [64,6,123,123] -> [64,15,62,62] (k=3, s=2, p=1)
// =============================================================
__global__ __launch_bounds__(256) void conv2_relu_kernel(
    const float* __restrict__ in, const float* __restrict__ w,
    const float* __restrict__ bias, float* __restrict__ out)
{
    __shared__ float sw[15 * 6 * 9];   // 810
    __shared__ float sb[15];
    for (int i = threadIdx.x; i < 810; i += 256) sw[i] = w[i];
    if (threadIdx.x < 15) sb[threadIdx.x] = bias[threadIdx.x];
    __syncthreads();

    int tid = blockIdx.x * 256 + threadIdx.x;
    if (tid >= SZ_CONV2) return;
    int ox = tid % 62; int t = tid / 62;
    int oy = t % 62;   t /= 62;
    int oc = t % 15;   int b = t / 15;

    const float* inb = in + (b * 6) * 123 * 123;
    const float* wf  = sw + oc * 54;
    float acc = sb[oc];
    int iy0 = oy * 2 - 1, ix0 = ox * 2 - 1;
#pragma unroll
    for (int ic = 0; ic < 6; ++ic) {
        const float* inc = inb + ic * 123 * 123;
#pragma unroll
        for (int ky = 0; ky < 3; ++ky) {
            int iy = iy0 + ky;
            if (iy < 0 || iy >= 123) continue;
            const float* row = inc + iy * 123;
#pragma unroll
            for (int kx = 0; kx < 3; ++kx) {
                int ix = ix0 + kx;
                if (ix >= 0 && ix < 123)
                    acc = fmaf(row[ix], wf[ic * 9 + ky * 3 + kx], acc);
            }
        }
    }
    out[tid] = fmaxf(acc, 0.0f);
}

// =============================================================
// maxpool 2x2 s1 + flatten : [64,15,62,62] -> [64][LDX] (pad zeroed)
// =============================================================
__global__ __launch_bounds__(256) void pool2_flatten_kernel(
    const float* __restrict__ in, float* __restrict__ out)
{
    int tid = blockIdx.x * 256 + threadIdx.x;
    if (tid < BATCH) out[tid * LDX + K1] = 0.0f;   // zero the K-pad element
    if (tid >= BATCH * K1) return;
    int x = tid % 61; int t = tid / 61;
    int y = t % 61;   t /= 61;
    int c = t % 15;   int b = t / 15;
    const float* p = in + ((b * 15 + c) * 62 + y) * 62 + x;
    float m = fmaxf(fmaxf(p[0], p[1]), fmaxf(p[62], p[63]));
    out[b * LDX + c * 3721 + y * 61 + x] = m;
}

// =============================================================
// fc1 split-K partials: P[z][64][120] = X[:,kz] @ W[:,kz]^T
// fp32 WMMA 16x16x4; grid (4 mtiles, 8 ntiles, SPLITS); 8 waves split
// each z-chunk further.  Main loop is guard-free (A rows are zero-padded,
// B K-tail peeled to a wave-uniform epilogue owned by one wave).
// A frag (16x4): lanes 0-15 hold K=k0,k0+1; lanes 16-31 hold K=k0+2,k0+3.
// B frag (4x16): same K split, N = lane&15.
// =============================================================
__global__ __launch_bounds__(256) void fc1_wmma_kernel(
    const float* __restrict__ X, const float* __restrict__ W,
    float* __restrict__ P)
{
    __shared__ float red[8 * 256];
    const int lane = threadIdx.x & 31;
    const int wave = threadIdx.x >> 5;
    const int m0 = blockIdx.x * 16;
    const int n0 = blockIdx.y * 16;
    const int z  = blockIdx.z;
    const int hi  = lane >> 4;      // K half select
    const int l15 = lane & 15;

    const int arow  = m0 + l15;
    const int ncol  = n0 + l15;
    const float nmask = (ncol < N1) ? 1.0f : 0.0f;
    const int nrow  = (ncol < N1) ? ncol : 0;

    const int CH      = (S_STEPS + SPLITS - 1) / SPLITS;      // 250 steps/chunk
    const int s_begin = z * CH;
    const int s_end   = (s_begin + CH < S_STEPS) ? s_begin + CH : S_STEPS;
    const int s_full  = (s_end < S_STEPS - 1) ? s_end : S_STEPS - 1; // guard-free

    const float* Ap = X + (long long)arow * LDX + hi * 2;   // 8B aligned (LDX%4==0)
    const float* Bp = W + (long long)nrow * K1  + hi * 2;

    const float* a_ptr = Ap + (long long)(s_begin + wave) * 4;
    const float* b_ptr = Bp + (long long)(s_begin + wave) * 4;

    v8f acc = {0.f, 0.f, 0.f, 0.f, 0.f, 0.f, 0.f, 0.f};
    int s = s_begin + wave;
#pragma unroll 2
    for (; s < s_full; s += 8) {
        __builtin_prefetch(b_ptr + 256, 0, 1);              // global_prefetch_b8
        v2f a = *(const v2f*)a_ptr;
        v2f b;
        b.x = b_ptr[0] * nmask;
        b.y = b_ptr[1] * nmask;
        acc = __builtin_amdgcn_wmma_f32_16x16x4_f32(
            false, a, false, b, (short)0, acc, false, false);
        a_ptr += 32;                                        // 8 waves * 4 K
        b_ptr += 32;
    }
    if (s == S_STEPS - 1) {          // one wave of the last z-block: K tail
        v2f a = *(const v2f*)a_ptr;  // safe: row pad element is zero
        int kk = s * 4 + hi * 2;
        float b0 = (kk     < K1) ? b_ptr[0] : 0.0f;
        float b1 = (kk + 1 < K1) ? b_ptr[1] : 0.0f;
        v2f b; b.x = b0 * nmask; b.y = b1 * nmask;
        acc = __builtin_amdgcn_wmma_f32_16x16x4_f32(
            false, a, false, b, (short)0, acc, false, false);
    }

    // cross-wave reduction through LDS, then emit this split's partial tile
#pragma unroll
    for (int r = 0; r < 8; ++r) red[(wave * 32 + lane) * 8 + r] = acc[r];
    __syncthreads();

    int l = threadIdx.x >> 3;        // 0..31 : output lane
    int r = threadIdx.x & 7;         // 0..7  : output vgpr
    float ssum = 0.0f;
#pragma unroll
    for (int w = 0; w < 8; ++w) ssum += red[(w * 32 + l) * 8 + r];
    int row = m0 + r + ((l >> 4) ? 8 : 0);
    int col = n0 + (l & 15);
    if (col < N1) P[(z * BATCH + row) * N1 + col] = ssum;
}

// =============================================================
// fc1 reduction over splits + bias + ReLU -> Y[64,120]
// =============================================================
__global__ __launch_bounds__(256) void fc1_reduce_kernel(
    const float* __restrict__ P, const float* __restrict__ bias,
    float* __restrict__ Y)
{
    int t = blockIdx.x * 256 + threadIdx.x;
    if (t >= BATCH * N1) return;
    int col = t % N1;
    float s = bias[col];
#pragma unroll 8
    for (int z = 0; z < SPLITS; ++z) s += P[z * BATCH * N1 + t];
    Y[t] = fmaxf(s, 0.0f);
}

// =============================================================
// fc2: Y[64,84] = relu(X[64,120] @ W[84,120]^T + b), one wave per tile
// =============================================================
__global__ __launch_bounds__(32) void fc2_wmma_kernel(
    const float* __restrict__ X, const float* __restrict__ W,
    const float* __restrict__ bias, float* __restrict__ Y)
{
    const int lane = threadIdx.x & 31;
    const int m0 = blockIdx.x * 16;
    const int n0 = blockIdx.y * 16;
    const int hi  = lane >> 4;
    const int l15 = lane & 15;

    const int arow = m0 + l15;
    const int ncol = n0 + l15;
    const float nmask = (ncol < N2) ? 1.0f : 0.0f;
    const int nrow = (ncol < N2) ? ncol : 0;

    const float* Ap = X + arow * K2 + hi * 2;
    const float* Bp = W + nrow * K2 + hi * 2;

    v8f acc = {0.f, 0.f, 0.f, 0.f, 0.f, 0.f, 0.f, 0.f};
#pragma unroll 5
    for (int k0 = 0; k0 < K2; k0 += 4) {
        v2f a = *(const v2f*)(Ap + k0);
        v2f b = *(const v2f*)(Bp + k0);
        b.x *= nmask; b.y *= nmask;
        acc = __builtin_amdgcn_wmma_f32_16x16x4_f32(
            false, a, false, b, (short)0, acc, false, false);
    }
#pragma unroll
    for (int r = 0; r < 8; ++r) {
        int row = m0 + r + hi * 8;
        int col = n0 + l15;
        if (col < N2) Y[row * N2 + col] = fmaxf(acc[r] + bias[col], 0.0f);
    }
}

// =============================================================
// fc3 + exact single-qubit Bloch simulation + sigmoid head
// =============================================================
__global__ __launch_bounds__(64) void head_kernel(
    const float* __restrict__ X2, const float* __restrict__ w3,
    const float* __restrict__ b3, const float* __restrict__ q,
    float* __restrict__ out)
{
    int b = threadIdx.x;            // one lane per batch element (64 = 2 waves)
    const float* x = X2 + b * N2;
    float ang = b3[0];
#pragma unroll 4
    for (int j = 0; j < N2; ++j) ang = fmaf(x[j], w3[j], ang);

    float bx = sinf(ang), by = 0.0f, bz = cosf(ang);
#pragma unroll
    for (int d = 0; d < 3; ++d) {
        float phi = q[2 * d], th = q[2 * d + 1];
        float cp = cosf(phi), sp = sinf(phi);
        float tx = bx * cp - by * sp;
        by = bx * sp + by * cp;
        bx = tx;
        float ct = cosf(th), st = sinf(th);
        tx = bx * ct + bz * st;
        bz = bz * ct - bx * st;
        bx = tx;
    }
    const float SHIFT = 1.5707963267948966f;
    float p = 1.0f / (1.0f + expf(-(bz + SHIFT)));
    out[2 * b + 0] = p;
    out[2 * b + 1] = 1.0f - p;
}

// =============================================================
extern "C" void kernel_launch(void* const* d_in, const int* in_sizes, int n_in,
                              void* d_out, int out_size, void* d_ws, size_t ws_size,
                              hipStream_t stream) {
    const float* inp  = (const float*)d_in[0];
    const float* c1w  = (const float*)d_in[1];
    const float* c1b  = (const float*)d_in[2];
    const float* c2w  = (const float*)d_in[3];
    const float* c2b  = (const float*)d_in[4];
    const float* f1w  = (const float*)d_in[5];
    const float* f1b  = (const float*)d_in[6];
    const float* f2w  = (const float*)d_in[7];
    const float* f2b  = (const float*)d_in[8];
    const float* f3w  = (const float*)d_in[9];
    const float* f3b  = (const float*)d_in[10];
    const float* qp   = (const float*)d_in[11];

    float* bufA = (float*)d_ws;                 // conv1out -> conv2out -> P/Y/fc2out
    float* bufB = bufA + SZ_CONV1;              // pool1out -> flattened X (64 x LDX)

    conv1_relu_kernel<<<(SZ_CONV1 + 255) / 256, 256, 0, stream>>>(inp, c1w, c1b, bufA);
    pool1_kernel<<<(SZ_POOL1 + 255) / 256, 256, 0, stream>>>(bufA, bufB);
    conv2_relu_kernel<<<(SZ_CONV2 + 255) / 256, 256, 0, stream>>>(bufB, c2w, c2b, bufA);
    pool2_flatten_kernel<<<(BATCH * K1 + 255) / 256, 256, 0, stream>>>(bufA, bufB);

    float* P      = bufA;                       // [SPLITS][64][120] partials
    float* Y1     = P + SPLITS * BATCH * N1;    // [64][120]
    float* fc2out = Y1 + BATCH * N1;            // [64][84]

    fc1_wmma_kernel<<<dim3(BATCH / 16, (N1 + 15) / 16, SPLITS), 256, 0, stream>>>(bufB, f1w, P);
    fc1_reduce_kernel<<<(BATCH * N1 + 255) / 256, 256, 0, stream>>>(P, f1b, Y1);
    fc2_wmma_kernel<<<dim3(BATCH / 16, (N2 + 15) / 16), 32, 0, stream>>>(Y1, f2w, f2b, fc2out);
    head_kernel<<<1, 64, 0, stream>>>(fc2out, f3w, f3b, qp, (float*)d_out);
}